// RCNNTrainer_87943750352916
// MI455X (gfx1250) — compile-verified
//
#include <hip/hip_runtime.h>
#include <math.h>

// Problem constants (from reference)
#define BATCH 32
#define TGT   200
#define RPROP 4096
#define NCLS  81
#define ROWS_PER_BLOCK 8
#define NBLK ((BATCH * RPROP) / ROWS_PER_BLOCK)   // 16384 blocks, one wave32 per row
#define IOU_THRESH 0.7f
#define REDUCTION 16.0f

typedef __attribute__((ext_vector_type(2))) float v2f;
typedef __attribute__((ext_vector_type(8))) float v8f;

__device__ __forceinline__ float smooth_l1(float d) {
    float ad = fabsf(d);
    return (ad < 1.0f) ? 0.5f * ad * ad : ad - 0.5f;
}

// Kernel 1: one wave32 per proposal row (b, r).
// Computes best-IoU GT match, masked CE / accuracy / SmoothL1 contributions,
// then a deterministic per-block partial sum -> partials[block][4].
__global__ __launch_bounds__(256) void rcnn_row_kernel(
    const float* __restrict__ nms_reg,    // [B,R,4]
    const float* __restrict__ rcnn_reg,   // [B,R,4]
    const float* __restrict__ rcnn_cls,   // [B,R,C]
    const float* __restrict__ bboxes,     // [B,T,4]
    const int*   __restrict__ classes,    // [B,T]
    float* __restrict__ partials)         // [NBLK,4] : npos, ce, acc, reg
{
    __shared__ float sb[TGT * 4];
    __shared__ int   scls[TGT];
    __shared__ float wacc[ROWS_PER_BLOCK][4];

    const int tid  = threadIdx.x;
    const int lane = tid & 31;
    const int wid  = tid >> 5;
    const int row0 = blockIdx.x * ROWS_PER_BLOCK;
    const int b    = row0 / RPROP;          // whole block lies in one batch (4096 % 8 == 0)

    // Stage this batch's GT boxes + labels into LDS
    for (int i = tid; i < TGT * 4; i += 256) sb[i]   = bboxes[(size_t)b * TGT * 4 + i];
    for (int i = tid; i < TGT;     i += 256) scls[i] = classes[(size_t)b * TGT + i];
    __syncthreads();

    const int row = row0 + wid;             // == b*RPROP + r

    // Prefetch this row's logits (gfx1250 global_prefetch_b8)
    const float* lg = rcnn_cls + (size_t)row * NCLS;
    __builtin_prefetch(lg, 0, 3);

    // Proposal box (broadcast load; same address across the wave)
    const float4 a = *(const float4*)(nms_reg + (size_t)row * 4);
    const float area_a = (a.z - a.x) * (a.w - a.y);

    // ---- argmax IoU over T (first-occurrence tie-break) ----
    float bv = -INFINITY; int bi = 0x7fffffff;
    for (int t = lane; t < TGT; t += 32) {
        float gt = sb[t * 4 + 0], gl = sb[t * 4 + 1];
        float gb = sb[t * 4 + 2], gr = sb[t * 4 + 3];
        float it = fmaxf(a.x, gt), il = fmaxf(a.y, gl);
        float ib = fminf(a.z, gb), ir = fminf(a.w, gr);
        float inter  = fmaxf(ib - it, 0.0f) * fmaxf(ir - il, 0.0f);
        float area_g = (gb - gt) * (gr - gl);
        float iou = inter / (area_a + area_g - inter);
        if (iou > bv || (iou == bv && t < bi)) { bv = iou; bi = t; }
    }
    for (int off = 16; off; off >>= 1) {
        float ov = __shfl_xor(bv, off);
        int   oi = __shfl_xor(bi, off);
        if (ov > bv || (ov == bv && oi < bi)) { bv = ov; bi = oi; }
    }

    float posf = 0.0f, ce = 0.0f, accf = 0.0f, regv = 0.0f;
    if (bv > IOU_THRESH) {                  // wave-uniform branch
        posf = 1.0f;
        const int label = scls[bi];

        // Lane-strided view of the 81 logits: slots c = lane, lane+32, lane+64
        float v0 = lg[lane];
        float v1 = lg[lane + 32];
        bool  has2 = (lane + 64) < NCLS;
        float v2 = has2 ? lg[lane + 64] : -INFINITY;

        // wave max
        float m = fmaxf(fmaxf(v0, v1), v2);
        for (int off = 16; off; off >>= 1) m = fmaxf(m, __shfl_xor(m, off));

        // argmax (first occurrence) for accuracy
        float av = v0; int ac = lane;
        if (v1 > av)          { av = v1; ac = lane + 32; }
        if (has2 && v2 > av)  { av = v2; ac = lane + 64; }
        for (int off = 16; off; off >>= 1) {
            float ov = __shfl_xor(av, off);
            int   oc = __shfl_xor(ac, off);
            if (ov > av || (ov == av && oc < ac)) { av = ov; ac = oc; }
        }

        // logsumexp
        float se = expf(v0 - m) + expf(v1 - m) + (has2 ? expf(v2 - m) : 0.0f);
        for (int off = 16; off; off >>= 1) se += __shfl_xor(se, off);

        // logit at the label (label is wave-uniform)
        int slot = label >> 5;
        float sv = (slot == 0) ? v0 : ((slot == 1) ? v1 : v2);
        float tl = __shfl(sv, label & 31);

        ce   = m + logf(se) - tl;
        accf = (ac == label) ? 1.0f : 0.0f;

        // ---- regression target from matched GT box ----
        float g0 = sb[bi * 4 + 0], g1 = sb[bi * 4 + 1];
        float g2 = sb[bi * 4 + 2], g3 = sb[bi * 4 + 3];
        float r0 = rintf(g0 / REDUCTION) * REDUCTION;   // RNE == jnp.round
        float r1 = rintf(g1 / REDUCTION) * REDUCTION;
        float r2 = rintf(g2 / REDUCTION) * REDUCTION;
        float r3 = rintf(g3 / REDUCTION) * REDUCTION;
        float h = r2 - r0; float w = r3 - r1;
        h = (h == 0.0f) ? 1.0f : h;
        w = (w == 0.0f) ? 1.0f : w;
        float t0 = (g0 - r0) / h, t1 = (g1 - r1) / w;
        float t2 = (g2 - r2) / h, t3 = (g3 - r3) / w;
        const float4 pr = *(const float4*)(rcnn_reg + (size_t)row * 4);
        regv = smooth_l1(pr.x - t0) + smooth_l1(pr.y - t1)
             + smooth_l1(pr.z - t2) + smooth_l1(pr.w - t3);
    }

    // Deterministic per-block reduction (8 waves -> 4 floats, fixed order)
    if (lane == 0) {
        wacc[wid][0] = posf; wacc[wid][1] = ce;
        wacc[wid][2] = accf; wacc[wid][3] = regv;
    }
    __syncthreads();
    if (tid == 0) {
        float s0 = 0, s1 = 0, s2 = 0, s3 = 0;
        for (int k = 0; k < ROWS_PER_BLOCK; ++k) {
            s0 += wacc[k][0]; s1 += wacc[k][1];
            s2 += wacc[k][2]; s3 += wacc[k][3];
        }
        partials[(size_t)blockIdx.x * 4 + 0] = s0;
        partials[(size_t)blockIdx.x * 4 + 1] = s1;
        partials[(size_t)blockIdx.x * 4 + 2] = s2;
        partials[(size_t)blockIdx.x * 4 + 3] = s3;
    }
}

// Kernel 2: single wave32 reduces all block partials deterministically.
// Final cross-lane sum done with V_WMMA_F32_16X16X4_F32: A = ones(16x4),
// B = per-lane partial replicated into both B VGPRs. Since A is all-ones the
// result is layout-robust: every B element contributes once with weight 1.0
// and appears twice total, so sum over all 32 lanes of c[0] == 4 * true sum.
__global__ __launch_bounds__(32) void rcnn_reduce_kernel(
    const float* __restrict__ partials,   // [NBLK,4]
    float* __restrict__ out)              // [3] : cls_loss, reg_loss, acc
{
    const int lane = threadIdx.x;
    float acc0 = 0, acc1 = 0, acc2 = 0, acc3 = 0;
    for (int j = lane; j < NBLK; j += 32) {
        acc0 += partials[(size_t)j * 4 + 0];
        acc1 += partials[(size_t)j * 4 + 1];
        acc2 += partials[(size_t)j * 4 + 2];
        acc3 += partials[(size_t)j * 4 + 3];
    }
    float accq[4] = {acc0, acc1, acc2, acc3};

    v2f onesA; onesA.x = 1.0f; onesA.y = 1.0f;
    float S[4];
#pragma unroll
    for (int q = 0; q < 4; ++q) {
        v2f bq; bq.x = accq[q]; bq.y = accq[q];
        v8f c = {};
        c = __builtin_amdgcn_wmma_f32_16x16x4_f32(
                /*neg_a=*/false, onesA, /*neg_b=*/false, bq,
                /*c_mod=*/(short)0, c, /*reuse_a=*/false, /*reuse_b=*/false);
        float col = c[0];
        for (int off = 16; off; off >>= 1) col += __shfl_xor(col, off);
        S[q] = col * 0.25f;
    }

    if (lane == 0) {
        float npos  = S[0];
        bool  has   = npos > 0.0f;
        float denom = fmaxf(npos, 1.0f);
        out[0] = has ? S[1] / denom : 0.0f;   // cls_loss
        out[1] = has ? S[3] / denom : 0.0f;   // reg_loss
        out[2] = has ? S[2] / denom : 0.0f;   // accuracy
    }
}

extern "C" void kernel_launch(void* const* d_in, const int* in_sizes, int n_in,
                              void* d_out, int out_size, void* d_ws, size_t ws_size,
                              hipStream_t stream) {
    const float* nms_reg  = (const float*)d_in[0];
    // d_in[1] = nms_cls (unused by forward)
    const float* rcnn_reg = (const float*)d_in[2];
    const float* rcnn_cls = (const float*)d_in[3];
    const float* bboxes   = (const float*)d_in[4];
    const int*   classes  = (const int*)d_in[5];
    float* partials = (float*)d_ws;           // NBLK*4 floats = 256 KB scratch
    float* out      = (float*)d_out;

    rcnn_row_kernel<<<NBLK, 256, 0, stream>>>(nms_reg, rcnn_reg, rcnn_cls,
                                              bboxes, classes, partials);
    rcnn_reduce_kernel<<<1, 32, 0, stream>>>(partials, out);
}